// RecurrentCharLM_82128364634385
// MI455X (gfx1250) — compile-verified
//
#include <hip/hip_runtime.h>
#include <hip/hip_bf16.h>

typedef float v2f __attribute__((ext_vector_type(2)));
typedef float v8f __attribute__((ext_vector_type(8)));

#define VOCAB 96
#define HID   256
#define DEPTH 500
#define BATCH 512
#define SEQ   64

#define ROWS    16                 // batch rows per workgroup (one 16-row WMMA M-tile)
#define NWAVES  16                 // one wave per 16-col N-tile (16*16 = 256 = HID)
#define THREADS (NWAVES * 32)
#define NBLOCKS (BATCH / ROWS)     // 32 workgroups, fully independent

// LDS layout (floats), K-pair interleaved for conflict-free ds_load_b64:
//   h0 [0 .. 4096)    : h buffer 0 -> h[(k>>1)*32 + row*2 + (k&1)]   (16 KB)
//   h1 [4096 .. 8192) : h buffer 1                                   (16 KB)
// W lives entirely in VGPRs (64 x float2 B fragments per wave).
#define LDS_FLOATS (2 * 4096)      // 32 KB

__global__ __launch_bounds__(THREADS)
void RecurrentCharLM_82128364634385_kernel(
    const int*   __restrict__ chars,    // (BATCH, SEQ)
    const float* __restrict__ hidden,   // (BATCH, HID)
    const float* __restrict__ embed_w,  // (VOCAB, HID)
    const float* __restrict__ Wg,       // (HID, HID)   out[n] = sum_k h[k]*W[k][n]
    const float* __restrict__ ro_w,     // (VOCAB, HID)
    const float* __restrict__ ro_b,     // (VOCAB,)
    float* __restrict__ logits_out,     // (BATCH, SEQ, VOCAB)
    float* __restrict__ hfinal_out)     // (BATCH, HID)
{
    extern __shared__ float lds[];
    float* h0 = lds;
    float* h1 = lds + 4096;

    const int tid  = threadIdx.x;
    const int wave = tid >> 5;
    const int lane = tid & 31;
    const int half = lane >> 4;     // 0: lanes 0-15 hold {K,K+1}; 1: lanes 16-31 hold {K+2,K+3}
    const int lr   = lane & 15;
    const int r0   = blockIdx.x * ROWS;
    const int n0   = wave * 16;     // this wave's N-tile base column

    // ---- B fragments of W -> registers, once, for all 32,000 iterations ----
    // B vgpr0 = W[k0+2*half][n0+lr], vgpr1 = W[k0+1+2*half][n0+lr]
    v2f breg[HID / 4];
    {
        const float* wb = Wg + 2 * half * HID + (n0 + lr);
        #pragma unroll
        for (int k0 = 0; k0 < HID; k0 += 4) {
            v2f b;
            b.x = wb[(size_t)k0 * HID];
            b.y = wb[(size_t)(k0 + 1) * HID];
            breg[k0 >> 2] = b;
        }
    }

    // ---- Stage initial h tile (K-pair interleaved) ----
    for (int i = tid; i < ROWS * 128; i += THREADS) {
        const int row = i >> 7, kk = i & 127;
        v2f hv = *(const v2f*)(hidden + (r0 + row) * HID + 2 * kk);
        *(v2f*)(h0 + kk * 32 + row * 2) = hv;
    }
    // Warm ro_w (96 KB) into cache for the per-timestep readout
    for (int i = tid; i < VOCAB * HID; i += THREADS * 16)
        __builtin_prefetch(ro_w + i, 0, 1);
    __syncthreads();

    // Per-lane constant bases: A fetch = base + (k0>>1)*32 floats (imm DS offset),
    // C store = base + j*2 floats.
    const float* aB0 = h0 + half * 32 + lr * 2;
    const float* aB1 = h1 + half * 32 + lr * 2;
    const int    n   = n0 + lr;
    const int    stIdx = (n >> 1) * 32 + (half * 8) * 2 + (n & 1);

    int cur = 0;

    for (int t = 0; t < SEQ; ++t) {
        // ---- h += embed_w[chars[:, t]]  (in place on current buffer) ----
        float* hc_w = cur ? h1 : h0;
        for (int i = tid; i < ROWS * HID; i += THREADS) {
            const int row = i >> 8, col = i & 255;
            const int c = chars[(r0 + row) * SEQ + t];
            hc_w[(col >> 1) * 32 + row * 2 + (col & 1)] += embed_w[c * HID + col];
        }
        __syncthreads();

        // ---- 500 x { h = ReLU(h @ W) } : latency-critical chain ----
        for (int it = 0; it < DEPTH; ++it) {
            const float* aB = cur ? aB1 : aB0;
            float*       hn = cur ? h0 : h1;

            v8f acc = {};
            #pragma unroll
            for (int k0 = 0; k0 < HID; k0 += 4) {
                v2f a = *(const v2f*)(aB + (k0 >> 1) * 32);   // ds_load_b64, imm offset
                acc = __builtin_amdgcn_wmma_f32_16x16x4_f32(
                          false, a, false, breg[k0 >> 2], (short)0, acc, false, false);
            }
            // ReLU + scatter into packed next-h layout (imm-offset b32 stores)
            #pragma unroll
            for (int j = 0; j < 8; ++j) {
                const float v = acc[j] > 0.0f ? acc[j] : 0.0f;
                hn[stIdx + j * 2] = v;
            }
            __syncthreads();
            cur ^= 1;
        }
        // DEPTH is even -> cur parity unchanged across the loop

        // ---- readout: logits[:, t] = h @ ro_w.T + ro_b  (waves 0..5 cover 96 vocab) ----
        if (wave < 6) {
            const float* aB = cur ? aB1 : aB0;
            const float* rp = ro_w + (n0 + lr) * HID + half * 2;
            v8f acc = {};
            #pragma unroll
            for (int k0 = 0; k0 < HID; k0 += 4) {
                v2f a = *(const v2f*)(aB + (k0 >> 1) * 32);
                v2f b; b.x = rp[k0]; b.y = rp[k0 + 1];        // B[k][n] = ro_w[n][k]
                acc = __builtin_amdgcn_wmma_f32_16x16x4_f32(
                          false, a, false, b, (short)0, acc, false, false);
            }
            const float bias = ro_b[n0 + lr];
            #pragma unroll
            for (int j = 0; j < 8; ++j) {
                const int row = r0 + j + half * 8;
                logits_out[(row * SEQ + t) * VOCAB + n0 + lr] = acc[j] + bias;
            }
        }
        __syncthreads();   // readout reads cur; next t's embed writes cur
    }

    // ---- h_final ----
    {
        const float* hc = cur ? h1 : h0;
        for (int i = tid; i < ROWS * HID; i += THREADS) {
            const int row = i >> 8, col = i & 255;
            hfinal_out[(r0 + row) * HID + col] =
                hc[(col >> 1) * 32 + row * 2 + (col & 1)];
        }
    }
}

extern "C" void kernel_launch(void* const* d_in, const int* in_sizes, int n_in,
                              void* d_out, int out_size, void* d_ws, size_t ws_size,
                              hipStream_t stream) {
    const int*   chars   = (const int*)  d_in[0];
    const float* hidden  = (const float*)d_in[1];
    const float* embed_w = (const float*)d_in[2];
    const float* Wg      = (const float*)d_in[3];
    const float* ro_w    = (const float*)d_in[4];
    const float* ro_b    = (const float*)d_in[5];

    float* logits_out = (float*)d_out;                               // (512,64,96)
    float* hfinal_out = (float*)d_out + (size_t)BATCH * SEQ * VOCAB; // (512,256)

    RecurrentCharLM_82128364634385_kernel
        <<<NBLOCKS, THREADS, LDS_FLOATS * sizeof(float), stream>>>(
            chars, hidden, embed_w, Wg, ro_w, ro_b, logits_out, hfinal_out);
}